// CrossHeadProjectionV2_74998718923392
// MI455X (gfx1250) — compile-verified
//
#include <hip/hip_runtime.h>

// CDNA5 / gfx1250 — wave32, fp32 WMMA path (V_WMMA_F32_16X16X4_F32).
typedef float v2f __attribute__((ext_vector_type(2)));
typedef float v8f __attribute__((ext_vector_type(8)));

#define T_DIM 2048
#define S_DIM 2048
#define TS    (T_DIM * S_DIM)

// ds_swizzle SWAPX16: group-of-32, xor_mask=0x10, or=0, and=0x1f -> 0x401f
__device__ __forceinline__ float swap16(float v) {
    return __int_as_float(__builtin_amdgcn_ds_swizzle(__float_as_int(v), 0x401f));
}

// ---------------------------------------------------------------------------
// Kernel 1: fold static mix + query-side low-rank + qdd diagonal into one
// 16x16 matrix per t:  Qt[m][n] = w[m][n] + (m==n)*(1+qdd[t][m])
//                               + sum_i qw1[t][i][m]*qw2[t][i][n]
// Stored directly in WMMA A-operand order so the main kernel fetches each
// lane's 8 A values with two b128 loads:
//   qtA[t*256 + (m>>3)*128 + n*8 + (m&7)]
// (m = 8*hi + 2*mc + j  ->  lane(hi, n=lo) reads 8 contiguous floats.)
// 2 MB total -> lives in L2 for the whole main kernel.
// ---------------------------------------------------------------------------
__global__ __launch_bounds__(256) void precompute_qt_kernel(
    const float* __restrict__ w,   const float* __restrict__ qw1,
    const float* __restrict__ qw2, const float* __restrict__ qdd,
    float* __restrict__ qtA)
{
    const int t  = blockIdx.x;
    const int mn = threadIdx.x;     // m*16 + n
    const int m  = mn >> 4;
    const int n  = mn & 15;
    float v = w[m * 16 + n];
    if (m == n) v += 1.0f + qdd[t * 16 + m];
    v += qw1[t * 32 +      m] * qw2[t * 32 +      n];   // i = 0
    v += qw1[t * 32 + 16 + m] * qw2[t * 32 + 16 + n];   // i = 1
    qtA[t * 256 + (m >> 3) * 128 + n * 8 + (m & 7)] = v;
}

// ---------------------------------------------------------------------------
// Kernel 2: main streaming kernel.
//   out[n,t,s] = sum_m Qt[t][m][n] * x[m,t,s]      (4x v_wmma_f32_16x16x4_f32)
//              + sum_i h_i(s) * kw2[s][i][n]       (rank-2, packed VALU)
//              + kdd[s][n] * x[n,t,s]              (diag, own register)
//   with h_i(s) = sum_m kw1[s][i][m] * x[m,t,s]
//
// Contraction-index permutation: chunk mc, k_local kl carries global head
//   m = 8*(kl>>1) + 2*mc + (kl&1)
// With the ISA operand layouts (lane half hi holds kl = j + 2*hi), each
// lane's own B registers are exactly x[m = 8*hi + 2*mc + j][s] — the lane's
// OWN output-row range [8*hi, 8*hi+8):
//   * diagonal term:  x[v+8*hi][s] == b[v>>1][v&1]   (no shuffle, no select)
//   * h_i: 8-FMA partial over own rows + one SWAPX16 swizzle + add
// All loop-body accesses are base+immediate (row bases hoisted; t-stride
// 8*S*4 = 64KB per iteration fits the signed 24-bit instruction offset).
// ---------------------------------------------------------------------------
__global__ __launch_bounds__(256) void chp_main_kernel(
    const float* __restrict__ x,   const float* __restrict__ kw1,
    const float* __restrict__ kw2, const float* __restrict__ kdd,
    const float* __restrict__ qtA, float* __restrict__ out)
{
    const int wave = threadIdx.x >> 5;
    const int lane = threadIdx.x & 31;
    const int lo   = lane & 15;          // column within tile / n low bits
    const int hi   = lane >> 4;          // half-wave select
    const int s0   = blockIdx.x * 16;    // s-tile base
    const int s    = s0 + lo;            // this lane's column

    // ---- per-lane key-side weights for column s (registers, reused over t) --
    float kw1r[2][8];    // kw1[s][i][8*hi + r] — multiplies own b values
    float kw2r[2][8];    // kw2[s][i][n], n = v + 8*hi
    float kddr[8];       // kdd[s][n],    n = v + 8*hi
    {
        #pragma unroll
        for (int i = 0; i < 2; ++i) {
            const float4* p1 = (const float4*)(kw1 + (size_t)s * 32 + i * 16 + hi * 8);
            float4 u0 = p1[0], u1 = p1[1];
            kw1r[i][0] = u0.x; kw1r[i][1] = u0.y; kw1r[i][2] = u0.z; kw1r[i][3] = u0.w;
            kw1r[i][4] = u1.x; kw1r[i][5] = u1.y; kw1r[i][6] = u1.z; kw1r[i][7] = u1.w;
            const float4* p2 = (const float4*)(kw2 + (size_t)s * 32 + i * 16 + hi * 8);
            float4 v0 = p2[0], v1 = p2[1];
            kw2r[i][0] = v0.x; kw2r[i][1] = v0.y; kw2r[i][2] = v0.z; kw2r[i][3] = v0.w;
            kw2r[i][4] = v1.x; kw2r[i][5] = v1.y; kw2r[i][6] = v1.z; kw2r[i][7] = v1.w;
        }
        const float4* pd = (const float4*)(kdd + (size_t)s * 16 + hi * 8);
        float4 d0 = pd[0], d1 = pd[1];
        kddr[0] = d0.x; kddr[1] = d0.y; kddr[2] = d0.z; kddr[3] = d0.w;
        kddr[4] = d1.x; kddr[5] = d1.y; kddr[6] = d1.z; kddr[7] = d1.w;
    }

    const int tbase = blockIdx.y * 64 + wave;   // 8 waves, stride-8 t coverage

    // ---- hoisted base pointers: all loop-body accesses are base + imm ----
    const float* xrow[8];   // x  row m = 8*hi + r, at (tbase, s0+lo)
    float*       orow[8];   // out row n = 8*hi + r, at (tbase, s0+lo)
    #pragma unroll
    for (int r = 0; r < 8; ++r) {
        xrow[r] = x   + (size_t)(8 * hi + r) * TS + (size_t)tbase * S_DIM + s0 + lo;
        orow[r] = out + (size_t)(8 * hi + r) * TS + (size_t)tbase * S_DIM + s0 + lo;
    }
    // A-operand base: lane's 8 contiguous Qt values per t (two b128 loads)
    const float* qtb = qtA + (size_t)tbase * 256 + hi * 128 + lo * 8;

    #pragma unroll
    for (int it = 0; it < 8; ++it) {
        const int toff = it * 8 * S_DIM;     // x/out immediate (elements)
        const int qoff = it * 8 * 256;       // qtA immediate (elements)

        // ---- load WMMA operands (all coalesced; Qt hits L2) ----
        const float4 q0 = *(const float4*)(qtb + qoff);
        const float4 q1 = *(const float4*)(qtb + qoff + 4);
        v2f a[4], b[4];
        a[0][0] = q0.x; a[0][1] = q0.y;
        a[1][0] = q0.z; a[1][1] = q0.w;
        a[2][0] = q1.x; a[2][1] = q1.y;
        a[3][0] = q1.z; a[3][1] = q1.w;
        #pragma unroll
        for (int mc = 0; mc < 4; ++mc) {
            b[mc][0] = xrow[2 * mc + 0][toff];
            b[mc][1] = xrow[2 * mc + 1][toff];
        }

        // ---- 4 chained fp32 WMMAs: Y += Qt^T(chunk) * X(chunk) ----
        v8f c = {0.f, 0.f, 0.f, 0.f, 0.f, 0.f, 0.f, 0.f};
        #pragma unroll
        for (int mc = 0; mc < 4; ++mc) {
            c = __builtin_amdgcn_wmma_f32_16x16x4_f32(
                    /*neg_a=*/false, a[mc], /*neg_b=*/false, b[mc],
                    /*c_mod=*/(short)0, c, /*reuse_a=*/false, /*reuse_b=*/false);
        }

        // ---- key-side rank-2: partial over own rows, complete via SWAPX16 --
        float ph0 = 0.f, ph1 = 0.f;
        #pragma unroll
        for (int mc = 0; mc < 4; ++mc) {
            #pragma unroll
            for (int j = 0; j < 2; ++j) {
                ph0 += kw1r[0][2 * mc + j] * b[mc][j];
                ph1 += kw1r[1][2 * mc + j] * b[mc][j];
            }
        }
        const float h0 = ph0 + swap16(ph0);
        const float h1 = ph1 + swap16(ph1);

        // ---- accumulate rank-2 + diagonal; x[v+8*hi][s] is own register ----
        #pragma unroll
        for (int v = 0; v < 8; ++v) {
            c[v] += h0 * kw2r[0][v] + h1 * kw2r[1][v] + kddr[v] * b[v >> 1][v & 1];
        }

        // ---- store D layout: lane writes rows n = v + 8*hi, col s0+lo ----
        #pragma unroll
        for (int v = 0; v < 8; ++v) {
            orow[v][toff] = c[v];
        }
    }
}

// ---------------------------------------------------------------------------
// Launch: inputs (setup_inputs order): inputs, w, qw1, qw2, kw1, kw2, qdd, kdd
// Workspace: needs T*256*4 = 2 MB for the precomputed per-t matrices.
// ---------------------------------------------------------------------------
extern "C" void kernel_launch(void* const* d_in, const int* in_sizes, int n_in,
                              void* d_out, int out_size, void* d_ws, size_t ws_size,
                              hipStream_t stream) {
    const float* x   = (const float*)d_in[0];
    const float* w   = (const float*)d_in[1];
    const float* qw1 = (const float*)d_in[2];
    const float* qw2 = (const float*)d_in[3];
    const float* kw1 = (const float*)d_in[4];
    const float* kw2 = (const float*)d_in[5];
    const float* qdd = (const float*)d_in[6];
    const float* kdd = (const float*)d_in[7];
    float* out = (float*)d_out;
    float* qtA = (float*)d_ws;   // T*256 floats = 2 MB

    precompute_qt_kernel<<<T_DIM, 256, 0, stream>>>(w, qw1, qw2, qdd, qtA);

    dim3 grid(S_DIM / 16, T_DIM / 64);   // 128 s-tiles x 32 t-blocks
    chp_main_kernel<<<grid, 256, 0, stream>>>(x, kw1, kw2, kdd, qtA, out);
}